// CrossCorrelLoss_43155831390616
// MI455X (gfx1250) — compile-verified
//
#include <hip/hip_runtime.h>

typedef __attribute__((ext_vector_type(2))) float v2f;
typedef __attribute__((ext_vector_type(8))) float v8f;

#define N_FEAT 321
#define NP 352                        // 11 * 32 padded feature dim
#define MROWS 65536                   // B*T rows
#define KSPLIT 64
#define KCHUNK (MROWS / KSPLIT)       // 1024
#define NBLK 11                       // NP / 32
#define NTRIL (NBLK * (NBLK + 1) / 2) // 66 lower-tri 32x32 blocks
#define NTASKS (NTRIL * KSPLIT)       // 4224 wave tasks

// workspace float layout
#define WS_SUM 0
#define WS_SQ  352
#define WS_MU  704
#define WS_ISD 1056
#define WS_S   1408
#define WS_FLOATS (WS_S + NP * NP)    // 125312 floats (~490 KB)

__global__ void zero_ws_kernel(float* __restrict__ ws, float* __restrict__ out) {
    int i = blockIdx.x * blockDim.x + threadIdx.x;
    if (i < WS_FLOATS) ws[i] = 0.0f;
    if (i == 0) out[0] = 0.0f;
}

// Per-column sum and sum-of-squares; 256 rows per block, coalesced across 321 cols.
__global__ void colstats_kernel(const float* __restrict__ X, float* __restrict__ ws) {
    int t = threadIdx.x;
    if (t >= N_FEAT) return;
    const float* p = X + (size_t)blockIdx.x * 256u * N_FEAT + t;
    float s = 0.0f, s2 = 0.0f;
#pragma unroll 4
    for (int r = 0; r < 256; ++r) {
        float v = p[(size_t)r * N_FEAT];
        s += v;
        s2 += v * v;
    }
    atomicAdd(&ws[WS_SUM + t], s);
    atomicAdd(&ws[WS_SQ + t], s2);
}

__global__ void musd_kernel(float* __restrict__ ws) {
    int t = threadIdx.x;
    if (t >= N_FEAT) return;
    const float Mf = (float)MROWS;
    float mu  = ws[WS_SUM + t] / Mf;
    float var = (ws[WS_SQ + t] - Mf * mu * mu) / (Mf - 1.0f);  // ddof=1
    ws[WS_MU + t]  = mu;
    ws[WS_ISD + t] = 1.0f / sqrtf(var);
}

// S = X^T X (raw gram), lower-triangle 32x32 blocks, split-K, fp32 WMMA.
__global__ void __launch_bounds__(256) syrk_wmma_kernel(const float* __restrict__ X,
                                                        float* __restrict__ ws) {
    float* __restrict__ S = ws + WS_S;
    int wave = (blockIdx.x << 3) + (threadIdx.x >> 5);
    if (wave >= NTASKS) return;                  // wave-uniform: EXEC stays all-1s
    int tileId = wave / KSPLIT;
    int kc     = wave - tileId * KSPLIT;
    // decode lower-tri block (bi >= bj), row-major tril order
    int bi = (int)((sqrtf(8.0f * (float)tileId + 1.0f) - 1.0f) * 0.5f);
    while ((bi + 1) * (bi + 2) / 2 <= tileId) ++bi;
    while (bi * (bi + 1) / 2 > tileId) --bi;
    int bj = tileId - bi * (bi + 1) / 2;
    int rowBase = bi << 5;
    int colBase = bj << 5;

    int lane = threadIdx.x & 31;
    int m    = lane & 15;            // A/B column within 16-tile (M or N index)
    int kk   = (lane >> 4) << 1;     // half-wave selects K {0,1} vs {2,3}
    // clamp padded columns (>=321) to a safe address; those S entries are never read
    int ca0 = min(rowBase + m,      N_FEAT - 1);
    int ca1 = min(rowBase + 16 + m, N_FEAT - 1);
    int cb0 = min(colBase + m,      N_FEAT - 1);
    int cb1 = min(colBase + 16 + m, N_FEAT - 1);

    const float* p = X + (size_t)(kc * KCHUNK + kk) * N_FEAT;
    v8f acc00 = {}, acc01 = {}, acc10 = {}, acc11 = {};

    for (int s = 0; s < KCHUNK / 4; ++s) {
        v2f a0, a1, b0, b1;
        a0.x = p[ca0];          a0.y = p[N_FEAT + ca0];
        a1.x = p[ca1];          a1.y = p[N_FEAT + ca1];
        b0.x = p[cb0];          b0.y = p[N_FEAT + cb0];
        b1.x = p[cb1];          b1.y = p[N_FEAT + cb1];
        acc00 = __builtin_amdgcn_wmma_f32_16x16x4_f32(false, a0, false, b0, (short)0, acc00, false, false);
        acc01 = __builtin_amdgcn_wmma_f32_16x16x4_f32(false, a0, false, b1, (short)0, acc01, false, false);
        acc10 = __builtin_amdgcn_wmma_f32_16x16x4_f32(false, a1, false, b0, (short)0, acc10, false, false);
        acc11 = __builtin_amdgcn_wmma_f32_16x16x4_f32(false, a1, false, b1, (short)0, acc11, false, false);
        p += 4 * N_FEAT;
    }

    // C/D layout: VGPR r -> M = r + 8*(lane>=16), N = lane&15
    int iloc = (lane >> 4) << 3;
    int jloc = lane & 15;
#pragma unroll
    for (int r = 0; r < 8; ++r) {
        int i0 = rowBase + iloc + r;
        int j0 = colBase + jloc;
        atomicAdd(&S[i0 * NP + j0],              acc00[r]);
        atomicAdd(&S[i0 * NP + j0 + 16],         acc01[r]);
        atomicAdd(&S[(i0 + 16) * NP + j0],       acc10[r]);
        atomicAdd(&S[(i0 + 16) * NP + j0 + 16],  acc11[r]);
    }
}

// G[i,j] = (S - M*mu_i*mu_j)/(M*sd_i*sd_j); loss = sum_{i>=j} |G - real| * 0.1
__global__ void loss_kernel(const float* __restrict__ real,
                            const float* __restrict__ ws,
                            float* __restrict__ out) {
    __shared__ float red[256];
    int tx = threadIdx.x, ty = threadIdx.y;
    int j = blockIdx.x * 16 + tx;
    int i = blockIdx.y * 16 + ty;
    float v = 0.0f;
    if (i < N_FEAT && j <= i) {
        const float* S = ws + WS_S;
        float mu_i  = ws[WS_MU + i],  mu_j  = ws[WS_MU + j];
        float isd_i = ws[WS_ISD + i], isd_j = ws[WS_ISD + j];
        const float Mf = (float)MROWS;
        float g = (S[i * NP + j] - Mf * mu_i * mu_j) * isd_i * isd_j * (1.0f / Mf);
        int k = (i * (i + 1)) / 2 + j;   // tril row-major index
        v = fabsf(g - real[k]);
    }
    int t = ty * 16 + tx;
    red[t] = v;
    __syncthreads();
    for (int off = 128; off > 0; off >>= 1) {
        if (t < off) red[t] += red[t + off];
        __syncthreads();
    }
    if (t == 0) atomicAdd(out, red[0] * 0.1f);
}

extern "C" void kernel_launch(void* const* d_in, const int* in_sizes, int n_in,
                              void* d_out, int out_size, void* d_ws, size_t ws_size,
                              hipStream_t stream) {
    const float* X    = (const float*)d_in[0];   // (128*512, 321) fp32
    const float* real = (const float*)d_in[1];   // (51681,) fp32
    float* out = (float*)d_out;                  // scalar fp32
    float* ws  = (float*)d_ws;

    zero_ws_kernel<<<(WS_FLOATS + 255) / 256, 256, 0, stream>>>(ws, out);
    colstats_kernel<<<MROWS / 256, 352, 0, stream>>>(X, ws);
    musd_kernel<<<1, 352, 0, stream>>>(ws);
    syrk_wmma_kernel<<<(NTASKS + 7) / 8, 256, 0, stream>>>(X, ws);
    loss_kernel<<<dim3(NP / 16, NP / 16), dim3(16, 16), 0, stream>>>(real, ws, out);
}